// TwoLevelFCNetwork_65962107732178
// MI455X (gfx1250) — compile-verified
//
#include <hip/hip_runtime.h>
#include <hip/hip_bf16.h>

// Problem constants (match reference): B=8192, D=256, H=512, O=64, K=8
#define NB    8192
#define ND    256
#define NH    512
#define NO    64
#define NK    8

#define MROWS 32                      // rows per block (2 WMMA row-tiles)
#define TILES_PER_EXPERT (NB / MROWS) // 256
#define XS_LD  260                    // 256 + 4 pad (row offset = 4 banks)
#define HS_LD  516                    // 512 + 4 pad

typedef __attribute__((ext_vector_type(2))) float v2f;
typedef __attribute__((ext_vector_type(8))) float v8f;

// ---------------------------------------------------------------------------
// Branch-free tanh: copysign(1 - 2/(exp(2*min(|x|,10))+1), x).
// v_exp_f32 + v_rcp_f32, no EXEC-divergent branches.
// ---------------------------------------------------------------------------
__device__ __forceinline__ float tanh_fast(float x) {
    float ax = fminf(fabsf(x), 10.0f);
    float e  = __expf(ax + ax);
    float r  = __builtin_amdgcn_rcpf(e + 1.0f);
    return copysignf(1.0f - (r + r), x);
}

// ---------------------------------------------------------------------------
// Phase 0: zero the per-expert counters
// ---------------------------------------------------------------------------
__global__ void zero_counts_kernel(int* __restrict__ counts) {
    if (threadIdx.x < NK) counts[threadIdx.x] = 0;
}

// ---------------------------------------------------------------------------
// Phase 1: hard one-hot routing; bucket row indices per expert.
// idx = mod(int(|x[b,7]*100|), 8)  (abs before int cast, trunc toward zero)
// ---------------------------------------------------------------------------
__global__ void route_kernel(const float* __restrict__ x,
                             int* __restrict__ counts,
                             int* __restrict__ perm) {
    int b = blockIdx.x * blockDim.x + threadIdx.x;
    if (b >= NB) return;
    float v = x[b * ND + 7];
    int idx = ((int)fabsf(v * 100.0f)) & (NK - 1);
    int slot = atomicAdd(counts + idx, 1);
    perm[idx * NB + slot] = b;
}

// ---------------------------------------------------------------------------
// One MLP layer for this wave: 32xKdim (LDS) @ KdimxNH (global) + bias, tanh.
// Each wave owns 2 row-tiles x 4 col-tiles; B fragments are reused across the
// two row-tiles (1:1 load-to-WMMA ratio). V_WMMA_F32_16X16X4_F32.
//   A layout: lane m = lane&15, k = vgpr + 2*(lane>>4).
//   B layout: lane n = lane&15, k = vgpr + 2*(lane>>4).
//   C/D layout: lane n = lane&15, row m = vgpr + 8*(lane>>4).
// ---------------------------------------------------------------------------
__device__ __forceinline__ void mlp_layer_tanh(
    const float* __restrict__ Als, int lda,           // LDS activations [32 x Kdim]
    const float* __restrict__ W,                      // global weights [Kdim x NH]
    const float* __restrict__ bias,                   // global bias [NH]
    float* __restrict__ Ols, int ldo,                 // LDS output [32 x NH]
    int Kdim, int wave, int lane)
{
    const int hi = lane >> 4;
    const int n  = lane & 15;

    v8f acc[2][4];                                    // [row-tile][col-tile]
    int col[4];
#pragma unroll
    for (int j = 0; j < 4; ++j) {
        col[j] = (wave * 4 + j) * 16 + n;
        float bv = bias[col[j]];                      // per-column bias -> lane scalar
#pragma unroll
        for (int e = 0; e < 8; ++e) { acc[0][j][e] = bv; acc[1][j][e] = bv; }
    }

    const float* arow0 = Als + n * lda + 2 * hi;              // rows 0..15
    const float* arow1 = Als + (16 + n) * lda + 2 * hi;       // rows 16..31
    for (int kk = 0; kk < Kdim; kk += 4) {
        v2f a0 = *(const v2f*)(arow0 + kk);           // ds_load_b64, bank-conflict-free
        v2f a1 = *(const v2f*)(arow1 + kk);
        const float* wrow = W + (kk + 2 * hi) * NH;
#pragma unroll
        for (int j = 0; j < 4; ++j) {
            v2f bb;
            bb[0] = wrow[col[j]];
            bb[1] = wrow[NH + col[j]];
            acc[0][j] = __builtin_amdgcn_wmma_f32_16x16x4_f32(
                false, a0, false, bb, (short)0, acc[0][j], false, false);
            acc[1][j] = __builtin_amdgcn_wmma_f32_16x16x4_f32(
                false, a1, false, bb, (short)0, acc[1][j], false, false);
        }
    }

#pragma unroll
    for (int rt = 0; rt < 2; ++rt)
#pragma unroll
        for (int j = 0; j < 4; ++j)
#pragma unroll
            for (int e = 0; e < 8; ++e)
                Ols[(rt * 16 + e + 8 * hi) * ldo + col[j]] = tanh_fast(acc[rt][j][e]);
}

// ---------------------------------------------------------------------------
// Phase 2: per-expert 32-row tiles through the 3-layer MLP.
// grid = NK * TILES_PER_EXPERT blocks, 256 threads (8 waves).
// Dynamic LDS: bufA[32*HS_LD] (Xs -> H1s), bufB[32*HS_LD] (H0s).
// ---------------------------------------------------------------------------
__global__ __launch_bounds__(256) void moe_expert_kernel(
    const float* __restrict__ x,
    const float* __restrict__ W0, const float* __restrict__ b0,
    const float* __restrict__ W1, const float* __restrict__ b1,
    const float* __restrict__ Wo, const float* __restrict__ bo,
    const int* __restrict__ counts, const int* __restrict__ perm,
    float* __restrict__ out)
{
    const int tile = blockIdx.x & (TILES_PER_EXPERT - 1);
    const int k    = blockIdx.x / TILES_PER_EXPERT;
    const int cnt  = counts[k];
    const int base = tile * MROWS;
    if (base >= cnt) return;                          // uniform per block

    extern __shared__ float smem[];
    float* bufA = smem;                               // Xs (ld XS_LD), later H1s (ld HS_LD)
    float* bufB = smem + MROWS * HS_LD;               // H0s (ld HS_LD)

    const int tid  = threadIdx.x;
    const int wave = tid >> 5;
    const int lane = tid & 31;
    const int hi   = lane >> 4;
    const int n    = lane & 15;
    const int* pk  = perm + k * NB;

    // ---- gather 32 rows of x into Xs (8 threads per row, float4 vectors) ----
    {
        int r   = tid >> 3;                           // 0..31 tile row
        int s   = base + r;
        int rid = pk[s < cnt ? s : cnt - 1];          // pad with duplicate of last row
        int cg  = (tid & 7) * 32;
        const float* src = x + (long)rid * ND + cg;
        float* dst = bufA + r * XS_LD + cg;
#pragma unroll
        for (int q = 0; q < 8; ++q)
            *(float4*)(dst + 4 * q) = *(const float4*)(src + 4 * q);
    }
    __syncthreads();

    // ---- layer 0: H0 = tanh(Xs[32x256] @ W0[k] + b0[k]) -> bufB ----
    mlp_layer_tanh(bufA, XS_LD,
                   W0 + (long)k * ND * NH, b0 + k * NH,
                   bufB, HS_LD, ND, wave, lane);
    __syncthreads();

    // ---- layer 1: H1 = tanh(H0[32x512] @ W1[k] + b1[k]) -> bufA (Xs dead) ----
    mlp_layer_tanh(bufB, HS_LD,
                   W1 + (long)k * NH * NH, b1 + k * NH,
                   bufA, HS_LD, NH, wave, lane);
    __syncthreads();

    // ---- layer 2: out = H1[32x512] @ Wo[k] + bo[k] ----
    // 2 row-tiles x 4 col-tiles = 8 tasks -> one per wave, full K, no reduction.
    {
        const int rt   = wave >> 2;                   // row-tile 0..1
        const int j    = wave & 3;                    // col-tile 0..3 (O=64)
        const int colo = j * 16 + n;
        float bv = bo[k * NO + colo];
        v8f acc;
#pragma unroll
        for (int e = 0; e < 8; ++e) acc[e] = bv;

        const float* hrow = bufA + (rt * 16 + n) * HS_LD + 2 * hi;
        const float* wb   = Wo + (long)k * NH * NO + colo;
        for (int kk = 0; kk < NH; kk += 4) {
            v2f a = *(const v2f*)(hrow + kk);
            v2f bb;
            bb[0] = wb[(kk + 2 * hi) * NO];
            bb[1] = wb[(kk + 2 * hi + 1) * NO];
            acc = __builtin_amdgcn_wmma_f32_16x16x4_f32(
                false, a, false, bb, (short)0, acc, false, false);
        }

        // scatter: row m = rt*16 + e + 8*hi; lanes 0..15 cover 64B contiguous
#pragma unroll
        for (int e = 0; e < 8; ++e) {
            int m   = rt * 16 + e + 8 * hi;
            int s   = base + m;
            int rid = pk[s < cnt ? s : cnt - 1];
            out[(long)rid * NO + colo] = acc[e];      // duplicates write identical bytes
        }
    }
}

// ---------------------------------------------------------------------------
// Launch: route -> compact -> expert GEMMs. All on `stream`, graph-safe.
// Workspace: counts[8] at offset 0, perm[8*8192] at offset 256 (~256.3 KB).
// ---------------------------------------------------------------------------
extern "C" void kernel_launch(void* const* d_in, const int* in_sizes, int n_in,
                              void* d_out, int out_size, void* d_ws, size_t ws_size,
                              hipStream_t stream) {
    const float* x  = (const float*)d_in[0];
    const float* W0 = (const float*)d_in[1];
    const float* b0 = (const float*)d_in[2];
    const float* W1 = (const float*)d_in[3];
    const float* b1 = (const float*)d_in[4];
    const float* Wo = (const float*)d_in[5];
    const float* bo = (const float*)d_in[6];
    float* out = (float*)d_out;

    int* counts = (int*)d_ws;
    int* perm   = (int*)((char*)d_ws + 256);

    zero_counts_kernel<<<1, 32, 0, stream>>>(counts);
    route_kernel<<<NB / 256, 256, 0, stream>>>(x, counts, perm);

    const size_t lds_bytes = 2u * MROWS * HS_LD * sizeof(float);  // ~129 KB
    moe_expert_kernel<<<NK * TILES_PER_EXPERT, 256, lds_bytes, stream>>>(
        x, W0, b0, W1, b1, Wo, bo, counts, perm, out);
}